// CrossTeacherAttention_87677462380783
// MI455X (gfx1250) — compile-verified
//
#include <hip/hip_runtime.h>
#include <math.h>

// Problem constants (from reference)
#define BB 8
#define CC 256
#define NN 1024   // H*W = 32*32
#define TT 3

typedef __attribute__((ext_vector_type(2))) float v2f;
typedef __attribute__((ext_vector_type(8))) float v8f;

// ---------------------------------------------------------------------------
// Kernel A: one 1x1-conv projection as a WMMA GEMM.
//   dst[b][n][o] = sum_c src[b][c][n] * Wm[o][c] + bias[o]
// Launched 7 times with direct pointers (no in-kernel pointer selection, so
// all memory ops lower to global_*, not flat_*).
// grid = (N/128, B), block = 256 (8 waves). Block computes a 128x256 output
// tile: X slice staged once in LDS, W^T staged in 64-channel chunks, each
// wave does 16 rows x 4 o-tiles per chunk with A-fragment reuse.
// ---------------------------------------------------------------------------
#define XPAD 136   // 2*XPAD % 64 == 16 -> lane halves hit disjoint banks
#define OCH  64    // o-chunk
#define WPAD 257   // per-o row stride: 257 % 64 == 1 -> conflict-free columns

__global__ __launch_bounds__(256) void proj_kernel(
    const float* __restrict__ src0,   // [B][C][N]
    const float* __restrict__ Wm,     // [Cout][Cin]
    const float* __restrict__ bias,   // [Cout]
    float* __restrict__ dst)          // [B][N][C]
{
    const int mtile = blockIdx.x;              // 0..7  (128 rows each)
    const int b     = blockIdx.y;
    const int mbase = mtile << 7;
    const float* src  = src0 + (size_t)b * CC * NN;
    float*       dstb = dst  + (size_t)b * NN * CC;

    extern __shared__ float plds[];
    float* Xs = plds;               // [256][XPAD]  Xs[k][m] = src[k][mbase+m]
    float* Ws = Xs + CC * XPAD;     // [OCH][WPAD]  Ws[oj][k] = Wm[oc+oj][k]

    const int tid = threadIdx.x;

    // Stage X tile (256 x 128) once, coalesced b128 both sides.
    for (int idx = tid; idx < CC * 32; idx += 256) {
        int k = idx >> 5, m4 = (idx & 31) << 2;
        *(float4*)(Xs + k * XPAD + m4) =
            *(const float4*)(src + (size_t)k * NN + mbase + m4);
    }

    const int wave = tid >> 5;
    const int lane = tid & 31;
    const int l15  = lane & 15;
    const int kof  = (lane >> 4) << 1;                 // A/B K sub-offset
    const int mloc = (wave << 4) + l15;                // A-frag local row
    const int rbase = (wave << 4) + ((lane >> 4) << 3);// C/D local row base

    for (int oc = 0; oc < CC; oc += OCH) {
        __syncthreads();  // Ws reuse + (first iter) X staging complete
        // Stage W^T chunk: coalesced global float4 reads, scalar LDS stores.
        for (int idx = tid; idx < OCH * 64; idx += 256) {
            int oj = idx >> 6, k4 = (idx & 63) << 2;
            float4 w4 = *(const float4*)(Wm + (size_t)(oc + oj) * CC + k4);
            float* wr = Ws + oj * WPAD + k4;
            wr[0] = w4.x; wr[1] = w4.y; wr[2] = w4.z; wr[3] = w4.w;
        }
        __syncthreads();

        v8f acc[4];
        #pragma unroll
        for (int i = 0; i < 4; ++i) acc[i] = (v8f){};

        const float* ar = Xs + mloc;
        for (int k0 = 0; k0 < CC; k0 += 4) {
            v2f a;
            // A frag (16x4 fp32): V0 = K0/K2 across lane halves, V1 = K1/K3
            a.x = ar[(k0 + kof)     * XPAD];
            a.y = ar[(k0 + kof + 1) * XPAD];
            #pragma unroll
            for (int ot = 0; ot < 4; ++ot) {
                const float* br = Ws + ((ot << 4) + l15) * WPAD + k0 + kof;
                v2f bf;
                bf.x = br[0];
                bf.y = br[1];
                acc[ot] = __builtin_amdgcn_wmma_f32_16x16x4_f32(
                              false, a, false, bf, (short)0, acc[ot],
                              false, false);
            }
        }

        #pragma unroll
        for (int ot = 0; ot < 4; ++ot) {
            const int ocol = oc + (ot << 4) + l15;
            const float bo = bias[ocol];
            #pragma unroll
            for (int i = 0; i < 8; ++i)
                dstb[(size_t)(mbase + rbase + i) * CC + ocol] = acc[ot][i] + bo;
        }
    }
}

// ---------------------------------------------------------------------------
// Kernel B: fused flash-attention per (teacher, batch, 64-query block).
// grid = (N/64, B, T), block = 128 (4 waves), dynamic LDS ~229 KB.
// Streaming softmax: running row-max M and row-sum L in LDS; logits
// round-trip through a padded LDS P-tile so P@V A-fragments have the right
// register layout. Stores O_t[b][n][c] = softmax(QK^T*s) @ V.
// ---------------------------------------------------------------------------
#define QROWS 64
#define KBLK  128
#define QPAD  260   // 256+4: conflict-free column reads, 16B-aligned rows
#define PPAD  132   // 128+4

__global__ __launch_bounds__(128) void attn_kernel(
    const float* __restrict__ Q, const float* __restrict__ K,
    const float* __restrict__ V, float* __restrict__ O)
{
    const int qb = blockIdx.x;   // 0..15
    const int b  = blockIdx.y;
    const int t  = blockIdx.z;
    const int qbase = qb << 6;

    extern __shared__ float lds[];
    float* Qs   = lds;                        // [64][QPAD]
    float* KVs  = Qs  + QROWS * QPAD;         // [128][QPAD]  (K then V)
    float* Ps   = KVs + KBLK  * QPAD;         // [64][PPAD]
    float* rowM = Ps  + QROWS * PPAD;         // [64]
    float* rowL = rowM + QROWS;               // [64]
    float* rowS = rowL + QROWS;               // [64]

    const float* Qg = Q + ((size_t)b * NN + qbase) * CC;
    const float* Kg = K + (size_t)(t * BB + b) * NN * CC;
    const float* Vg = V + (size_t)(t * BB + b) * NN * CC;

    const int tid = threadIdx.x;

    // Stage Q block (64 x 256)
    for (int idx = tid; idx < QROWS * 64; idx += 128) {
        int r = idx >> 6, c4 = (idx & 63) << 2;
        *(float4*)(Qs + r * QPAD + c4) = *(const float4*)(Qg + (size_t)r * CC + c4);
    }
    if (tid < QROWS) { rowM[tid] = -__builtin_inff(); rowL[tid] = 0.f; }

    const int wave = tid >> 5;
    const int lane = tid & 31;
    const int l15  = lane & 15;
    const int kof  = (lane >> 4) << 1;
    const int qw   = wave << 4;
    const int prow = qw + ((lane >> 4) << 3);

    v8f Oacc[16];
    #pragma unroll
    for (int i = 0; i < 16; ++i) Oacc[i] = (v8f){};

    const float scl = 0.0625f;  // C^-0.5

    for (int kb = 0; kb < NN / KBLK; ++kb) {
        __syncthreads();  // previous iteration done reading KVs
        const float* Kblk = Kg + (size_t)(kb << 7) * CC;
        for (int idx = tid; idx < KBLK * 64; idx += 128) {
            int r = idx >> 6, c4 = (idx & 63) << 2;
            *(float4*)(KVs + r * QPAD + c4) =
                *(const float4*)(Kblk + (size_t)r * CC + c4);
        }
        __syncthreads();

        // ---- S = (Q K^T) * scale : 8 key sub-tiles of 16, K-dim = 256 ----
        const float* arowQ = Qs + (qw + l15) * QPAD;
        #pragma unroll
        for (int kt = 0; kt < 8; ++kt) {
            const float* brow = KVs + ((kt << 4) + l15) * QPAD;
            v8f s = {};
            for (int k4 = 0; k4 < CC; k4 += 4) {
                v2f a, bf;
                a.x  = arowQ[k4 + kof]; a.y  = arowQ[k4 + kof + 1];
                bf.x = brow[k4 + kof];  bf.y = brow[k4 + kof + 1];
                s = __builtin_amdgcn_wmma_f32_16x16x4_f32(
                        false, a, false, bf, (short)0, s, false, false);
            }
            #pragma unroll
            for (int i = 0; i < 8; ++i)
                Ps[(prow + i) * PPAD + (kt << 4) + l15] = s[i] * scl;
        }
        __syncthreads();

        // ---- streaming softmax update (one thread per query row) ----
        if (tid < QROWS) {
            float* pr = Ps + tid * PPAD;
            float mx = rowM[tid];
            for (int k = 0; k < KBLK; ++k) mx = fmaxf(mx, pr[k]);
            float sc = __expf(rowM[tid] - mx);    // 0 on first block
            float sum = 0.f;
            for (int k = 0; k < KBLK; ++k) {
                float p = __expf(pr[k] - mx);
                pr[k] = p; sum += p;
            }
            rowL[tid] = rowL[tid] * sc + sum;
            rowM[tid] = mx;
            rowS[tid] = sc;
        }
        __syncthreads();

        // ---- rescale O accumulators, stage V block over KVs ----
        {
            float f[8];
            #pragma unroll
            for (int i = 0; i < 8; ++i) f[i] = rowS[prow + i];
            #pragma unroll
            for (int ct = 0; ct < 16; ++ct)
                #pragma unroll
                for (int i = 0; i < 8; ++i) Oacc[ct][i] *= f[i];
        }
        const float* Vblk = Vg + (size_t)(kb << 7) * CC;
        for (int idx = tid; idx < KBLK * 64; idx += 128) {
            int r = idx >> 6, c4 = (idx & 63) << 2;
            *(float4*)(KVs + r * QPAD + c4) =
                *(const float4*)(Vblk + (size_t)r * CC + c4);
        }
        if (kb + 1 < NN / KBLK)  // hint next K block toward L2 (global_prefetch_b8)
            __builtin_prefetch(Kg + (size_t)((kb + 1) << 7) * CC + tid * 16, 0, 1);
        __syncthreads();

        // ---- O += P @ V : 16 channel tiles, K-dim = 128 keys ----
        const float* arowP = Ps + (qw + l15) * PPAD;
        #pragma unroll
        for (int ct = 0; ct < 16; ++ct) {
            v8f o = Oacc[ct];
            for (int k4 = 0; k4 < KBLK; k4 += 4) {
                v2f a, bf;
                a.x = arowP[k4 + kof]; a.y = arowP[k4 + kof + 1];
                const float* bcol = KVs + (k4 + kof) * QPAD + (ct << 4) + l15;
                bf.x = bcol[0]; bf.y = bcol[QPAD];
                o = __builtin_amdgcn_wmma_f32_16x16x4_f32(
                        false, a, false, bf, (short)0, o, false, false);
            }
            Oacc[ct] = o;
        }
    }
    __syncthreads();

    // Normalize by softmax denominator and store O_t[b][n][c]
    {
        float inv[8];
        #pragma unroll
        for (int i = 0; i < 8; ++i) inv[i] = 1.0f / rowL[prow + i];
        float* Og = O + ((size_t)(t * BB + b) * NN + qbase) * CC;
        #pragma unroll
        for (int ct = 0; ct < 16; ++ct)
            #pragma unroll
            for (int i = 0; i < 8; ++i)
                Og[(size_t)(prow + i) * CC + (ct << 4) + l15] =
                    Oacc[ct][i] * inv[i];
    }
}

// ---------------------------------------------------------------------------
// Kernel C: out[b][c][n] = student[b][c][n] + mean_t O_t[b][n][c]
// (teacher softmax weights are exactly 1/3: each attn row sums to 1, so
//  attn.mean over keys is the constant 1/N for every teacher.)
// ---------------------------------------------------------------------------
__global__ __launch_bounds__(256) void combine_kernel(
    const float* __restrict__ student, const float* __restrict__ O,
    float* __restrict__ out)
{
    const int nt = blockIdx.x;   // N/32
    const int ct = blockIdx.y;   // C/32
    const int b  = blockIdx.z;
    const int tx = threadIdx.x & 31;
    const int ty = threadIdx.x >> 5;         // 0..7
    const int n0 = nt << 5, c0 = ct << 5;
    const size_t TSTRIDE = (size_t)BB * NN * CC;

    __shared__ float tile[32][33];
    #pragma unroll
    for (int r = 0; r < 4; ++r) {
        int n = n0 + ty + (r << 3);
        int c = c0 + tx;
        size_t base = ((size_t)b * NN + n) * CC + c;   // coalesced over c
        float s = O[base] + O[TSTRIDE + base] + O[2 * TSTRIDE + base];
        tile[ty + (r << 3)][tx] = s * (1.0f / 3.0f);
    }
    __syncthreads();
    #pragma unroll
    for (int r = 0; r < 4; ++r) {
        int c = c0 + ty + (r << 3);
        int n = n0 + tx;
        size_t idx = ((size_t)b * CC + c) * NN + n;    // coalesced over n
        out[idx] = student[idx] + tile[tx][ty + (r << 3)];
    }
}

// ---------------------------------------------------------------------------
extern "C" void kernel_launch(void* const* d_in, const int* in_sizes, int n_in,
                              void* d_out, int out_size, void* d_ws, size_t ws_size,
                              hipStream_t stream) {
    const float* student = (const float*)d_in[0];
    const float* tfeat[TT] = { (const float*)d_in[1], (const float*)d_in[2],
                               (const float*)d_in[3] };
    const float* Wq = (const float*)d_in[4];
    const float* bq = (const float*)d_in[5];
    const float* Wk = (const float*)d_in[6];
    const float* bk = (const float*)d_in[7];
    const float* Wv = (const float*)d_in[8];
    const float* bv = (const float*)d_in[9];

    float* ws = (float*)d_ws;
    float* Q = ws;                                   //  8 MB  [B][N][C]
    float* K = Q + (size_t)BB * NN * CC;             // 24 MB  [T][B][N][C]
    float* V = K + (size_t)TT * BB * NN * CC;        // 24 MB
    float* O = V + (size_t)TT * BB * NN * CC;        // 24 MB
    (void)ws_size; (void)in_sizes; (void)n_in; (void)out_size;

    // 1) projections: 7 launches, 64 blocks each (no in-kernel ptr selection)
    const size_t pLds = (size_t)(CC * XPAD + OCH * WPAD) * sizeof(float);
    hipFuncSetAttribute((const void*)proj_kernel,
                        hipFuncAttributeMaxDynamicSharedMemorySize, (int)pLds);
    proj_kernel<<<dim3(NN / 128, BB), 256, pLds, stream>>>(student, Wq, bq, Q);
    for (int t = 0; t < TT; ++t)
        proj_kernel<<<dim3(NN / 128, BB), 256, pLds, stream>>>(
            tfeat[t], Wk, bk, K + (size_t)t * BB * NN * CC);
    for (int t = 0; t < TT; ++t)
        proj_kernel<<<dim3(NN / 128, BB), 256, pLds, stream>>>(
            tfeat[t], Wv, bv, V + (size_t)t * BB * NN * CC);

    // 2) fused flash attention per (t,b,qblock), 384 blocks
    const size_t aLds = (size_t)(QROWS * QPAD + KBLK * QPAD + QROWS * PPAD +
                                 3 * QROWS) * sizeof(float);
    hipFuncSetAttribute((const void*)attn_kernel,
                        hipFuncAttributeMaxDynamicSharedMemorySize, (int)aLds);
    attn_kernel<<<dim3(NN / QROWS, BB, TT), 128, aLds, stream>>>(Q, K, V, O);

    // 3) fuse teachers (uniform 1/3) + residual, with LDS transpose
    combine_kernel<<<dim3(NN / 32, CC / 32, BB), 256, 0, stream>>>(
        student, O, (float*)d_out);
}